// ProdLayer_38706245271981
// MI455X (gfx1250) — compile-verified
//
#include <hip/hip_runtime.h>

// element_mars[nids] = node_mars[cids].sum(axis=1)
//   node_mars : [524288, 128] f32   (d_in[0])
//   element_mars (zeros, unused)    (d_in[1])
//   nids      : [N] int32           (d_in[2])   N = 262144
//   cids      : [N, 2] int32        (d_in[3])
//   out       : [N+1, 128] f32      (d_out)     row 0 must be zeros
//
// Memory-bound gather: ~387 MB total traffic -> ~16.6 us floor at 23.3 TB/s.
// One wave32 processes one 128-float row via one float4 (B128) per lane.
// 4 rows per wave amortizes the per-wave scalar preamble (~40 SALU insts)
// over 6 KB of traffic and keeps 8 B128 loads in flight per wave.

typedef float v4f __attribute__((ext_vector_type(4)));

#define BATCH_F4      32   // 128 floats per row = 32 float4 = one per lane
#define ROWS_PER_WAVE 4
#define THREADS       256  // 8 waves per block

__global__ __launch_bounds__(THREADS) void prod_layer_fwd(
    const float* __restrict__ node_mars,
    const int*   __restrict__ nids,
    const int*   __restrict__ cids,
    float*       __restrict__ out,
    int num_rows)
{
    const int lane = threadIdx.x & 31;
    const int wave = threadIdx.x >> 5;
    const int wavesPerBlock = blockDim.x >> 5;

    // Row 0 of the output is the reserved dummy slot: write zeros (out is
    // poisoned by the harness, and the reference leaves it zero).
    if (blockIdx.x == 0 && wave == 0) {
        v4f z = (v4f)0.0f;
        __builtin_nontemporal_store(z, ((v4f*)out) + lane);
    }

    const long long wid = (long long)blockIdx.x * wavesPerBlock + wave;
    // Wave-uniform base row; readfirstlane lets the compiler scalarize the
    // index loads (s_load via constant cache instead of 32 redundant lanes).
    const int row0 = __builtin_amdgcn_readfirstlane((int)(wid * ROWS_PER_WAVE));

    // Prefetch the index stream ahead of the dependent gathers
    // (global_prefetch_b8; indices are the head of the latency chain).
    __builtin_prefetch(&cids[2 * row0 + 2 * ROWS_PER_WAVE * 2 * THREADS / 32], 0, 0);

    const v4f* __restrict__ nm = (const v4f*)node_mars;
    v4f*       __restrict__ op = (v4f*)out;

    if (row0 + ROWS_PER_WAVE <= num_rows) {
        // ---- hot path: full tile, no per-row predicates ----
        int c[2 * ROWS_PER_WAVE];
        int n[ROWS_PER_WAVE];
#pragma unroll
        for (int r = 0; r < ROWS_PER_WAVE; ++r) {
            c[2 * r + 0] = cids[2 * (row0 + r) + 0];
            c[2 * r + 1] = cids[2 * (row0 + r) + 1];
            n[r]         = nids[row0 + r];
        }

        // Issue all 8 B128 gathers before consuming any.
        v4f a[ROWS_PER_WAVE], b[ROWS_PER_WAVE];
#pragma unroll
        for (int r = 0; r < ROWS_PER_WAVE; ++r) {
            a[r] = nm[(long long)c[2 * r + 0] * BATCH_F4 + lane];
            b[r] = nm[(long long)c[2 * r + 1] * BATCH_F4 + lane];
        }

        // Write-once output: non-temporal stores keep the 128 MB result from
        // evicting gather-reusable node_mars lines out of the 192 MB L2.
#pragma unroll
        for (int r = 0; r < ROWS_PER_WAVE; ++r) {
            __builtin_nontemporal_store(a[r] + b[r],
                                        op + (long long)n[r] * BATCH_F4 + lane);
        }
    } else {
        // ---- cold tail: per-row guarded (grid divides evenly in practice) ----
        for (int r = 0; r < ROWS_PER_WAVE; ++r) {
            const int row = row0 + r;
            if (row < num_rows) {
                const int c0 = cids[2 * row + 0];
                const int c1 = cids[2 * row + 1];
                const int n0 = nids[row];
                v4f va = nm[(long long)c0 * BATCH_F4 + lane];
                v4f vb = nm[(long long)c1 * BATCH_F4 + lane];
                __builtin_nontemporal_store(va + vb,
                                            op + (long long)n0 * BATCH_F4 + lane);
            }
        }
    }
}

extern "C" void kernel_launch(void* const* d_in, const int* in_sizes, int n_in,
                              void* d_out, int out_size, void* d_ws, size_t ws_size,
                              hipStream_t stream) {
    const float* node_mars = (const float*)d_in[0];
    // d_in[1] (element_mars) is all zeros; row 0 handled directly.
    const int* nids = (const int*)d_in[2];
    const int* cids = (const int*)d_in[3];
    float* out = (float*)d_out;

    const int num_rows = in_sizes[2];                 // 262144
    const int wavesPerBlock = THREADS / 32;           // 8
    const long long totalWaves =
        ((long long)num_rows + ROWS_PER_WAVE - 1) / ROWS_PER_WAVE;
    const int blocks = (int)((totalWaves + wavesPerBlock - 1) / wavesPerBlock);

    prod_layer_fwd<<<blocks, THREADS, 0, stream>>>(node_mars, nids, cids, out, num_rows);
}